// FreqAttentionLayer_20847771254796
// MI455X (gfx1250) — compile-verified
//
#include <hip/hip_runtime.h>
#include <math.h>

typedef __attribute__((ext_vector_type(2))) float v2f;
typedef __attribute__((ext_vector_type(4))) float v4f;
typedef __attribute__((ext_vector_type(8))) float v8f;

#define NSPLIT 8
#define NN     64
#define CC     512
#define HW     1024   // 32*32
#define CPART  64
#define HID    32

__constant__ float c_MX[8] = {0.f, 0.f, 1.f, -1.f, 0.f, 3.f, 0.f, 2.f};
__constant__ float c_MY[8] = {0.f, 1.f, 0.f, -1.f, 3.f, 0.f, 2.f, 0.f};

// -------------------------------------------------------------------------
// Pass 1: dct_freq[n,c] = sum_{h,w} x[n,c,h,w] * bh[h]*bw[w]   (bandwidth bound)
// One block = (n, 8 consecutive channels of one split). One wave per channel.
// Lane l, iteration k loads float4 at offset 32k + l  -> 512B contiguous per wave
// per load instruction (global_load_b128). Default RT temporal hint so x gets
// installed in the 192MB L2 for reuse by pass 3.
// -------------------------------------------------------------------------
__global__ void dct_pool_kernel(const float* __restrict__ x,
                                float* __restrict__ dct_freq) {
    __shared__ float bh[32];
    __shared__ float bw[32];

    const int b  = blockIdx.x;
    const int n  = b >> 6;            // 64 c-tiles per n
    const int c0 = (b & 63) * 8;      // 8 channels per block
    const int split = c0 >> 6;

    const int t = threadIdx.x;
    if (t < 64) {
        const int   pos  = t & 31;
        const float freq = (t < 32) ? c_MX[split] : c_MY[split];
        float r = cosf(3.14159265358979f * freq * (pos + 0.5f) / 32.0f) * 0.17677669529664f; // 1/sqrt(32)
        if (freq != 0.0f) r *= 1.41421356237310f;
        if (t < 32) bh[pos] = r; else bw[pos] = r;
    }
    __syncthreads();

    const int wave = t >> 5;
    const int lane = t & 31;
    const int c    = c0 + wave;

    const float4* __restrict__ row =
        (const float4*)(x + ((size_t)n * CC + c) * HW);

    const int w4 = lane & 7;   // which float4 within a 32-wide row
    const int hb = lane >> 3;  // base row 0..3

    const float b0 = bw[4 * w4 + 0];
    const float b1 = bw[4 * w4 + 1];
    const float b2 = bw[4 * w4 + 2];
    const float b3 = bw[4 * w4 + 3];

    float s = 0.0f;
#pragma unroll
    for (int k = 0; k < 8; ++k) {
        const int h = hb + 4 * k;
        const float4 v = row[h * 8 + w4];     // = row[32k + lane] across the wave
        s += bh[h] * (v.x * b0 + v.y * b1 + v.z * b2 + v.w * b3);
    }

    // wave32 reduction
#pragma unroll
    for (int off = 16; off > 0; off >>= 1)
        s += __shfl_xor(s, off, 32);

    if (lane == 0)
        dct_freq[(size_t)n * CC + c] = s;
}

// -------------------------------------------------------------------------
// Pass 2: gating network. One block per group g (8 blocks, 256 thr = 8 waves).
//  - m[n,g'] means + tiny MLP (VALU, trivially small)
//  - conv1: H = relu(G @ W1^T)   (64x64)@(64x32) via 8 WMMA f32 16x16x4 tiles
//  - conv2: att = sigmoid(H @ W2^T) (64x32)@(32x64) via 16 WMMA tiles
//  - scale[n, g*64+c] = att * fw[n]
// WMMA f32 16x16x4 operand layout (ISA 7.12.2):
//   A 16x4 : lanes 0-15 -> {K0,K1}, lanes 16-31 -> {K2,K3}, M = lane&15
//   B 4x16 : VGPR0 -> rows K0(low half)/K2(high half), VGPR1 -> K1/K3, N = lane&15
//   C/D    : VGPR j -> row (16*mt + j + 8*(lane>>4)), col = 16*nt + (lane&15)
// -------------------------------------------------------------------------
__global__ void gating_kernel(const float* __restrict__ dct_freq,
                              const float* __restrict__ W1,   // (8,32,64)
                              const float* __restrict__ W2,   // (8,64,32)
                              const float* __restrict__ M1,   // (8,8)
                              const float* __restrict__ M2,   // (8,8)
                              float* __restrict__ scale) {
    const int g = blockIdx.x;

    __shared__ float Gs[64][68];   // group's dct_freq slice, padded
    __shared__ float Hs[64][36];   // hidden activations, padded
    __shared__ float ms[64][8];    // per-split means
    __shared__ float fw[64];       // freq_weight[:, g]

    const int t = threadIdx.x;

    // stage group slice into LDS
    for (int e = t; e < 64 * 64; e += 256) {
        const int n = e >> 6, c = e & 63;
        Gs[n][c] = dct_freq[(size_t)n * CC + g * CPART + c];
    }
    // per-split means over all 8 splits (needed for the 8x8 MLP)
    for (int e = t; e < 64 * 8; e += 256) {
        const int n = e >> 3, gp = e & 7;
        const float* p = dct_freq + (size_t)n * CC + gp * CPART;
        float s = 0.0f;
        for (int c = 0; c < CPART; ++c) s += p[c];
        ms[n][gp] = s * (1.0f / 64.0f);
    }
    __syncthreads();

    if (t < 64) {
        const int n = t;
        float tt[8];
#pragma unroll
        for (int j = 0; j < 8; ++j) {
            float s = 0.0f;
#pragma unroll
            for (int gp = 0; gp < 8; ++gp) s += ms[n][gp] * M1[j * 8 + gp];
            tt[j] = fmaxf(s, 0.0f);
        }
        float s = 0.0f;
#pragma unroll
        for (int j = 0; j < 8; ++j) s += tt[j] * M2[g * 8 + j];
        fw[n] = 1.0f / (1.0f + __expf(-s));
    }
    __syncthreads();

    const int wave = t >> 5;
    const int lane = t & 31;
    const int half = lane >> 4;   // 0: K0/K1 pair, 1: K2/K3 pair
    const int l16  = lane & 15;

    const float* __restrict__ W1g = W1 + g * HID * CPART;  // [o][k]
    const float* __restrict__ W2g = W2 + g * CPART * HID;  // [c][o]

    // ---- conv1: 4 x 2 tiles of 16x16, one per wave, K=64 ----
    {
        const int mt = wave >> 1;
        const int nt = wave & 1;
        v8f acc = {};
#pragma unroll
        for (int k0 = 0; k0 < 64; k0 += 4) {
            const int ka = k0 + 2 * half;
            const int m  = mt * 16 + l16;
            const int o  = nt * 16 + l16;
            v2f a, bb;
            a.x  = Gs[m][ka];
            a.y  = Gs[m][ka + 1];
            bb.x = W1g[o * 64 + ka];       // B[k][o] = W1[g][o][k]
            bb.y = W1g[o * 64 + ka + 1];
            acc = __builtin_amdgcn_wmma_f32_16x16x4_f32(
                false, a, false, bb, (short)0, acc, false, false);
        }
#pragma unroll
        for (int j = 0; j < 8; ++j) {
            const int m = mt * 16 + j + 8 * half;
            const int o = nt * 16 + l16;
            Hs[m][o] = fmaxf(acc[j], 0.0f);   // relu
        }
    }
    __syncthreads();

    // ---- conv2: 4 x 4 tiles, two per wave, K=32 ----
    for (int tile = wave; tile < 16; tile += 8) {
        const int mt = tile >> 2;
        const int nt = tile & 3;
        v8f acc = {};
#pragma unroll
        for (int k0 = 0; k0 < 32; k0 += 4) {
            const int ka = k0 + 2 * half;
            const int m  = mt * 16 + l16;
            const int c  = nt * 16 + l16;
            v2f a, bb;
            a.x  = Hs[m][ka];
            a.y  = Hs[m][ka + 1];
            bb.x = W2g[c * 32 + ka];       // B[o][c] = W2[g][c][o]
            bb.y = W2g[c * 32 + ka + 1];
            acc = __builtin_amdgcn_wmma_f32_16x16x4_f32(
                false, a, false, bb, (short)0, acc, false, false);
        }
#pragma unroll
        for (int j = 0; j < 8; ++j) {
            const int n = mt * 16 + j + 8 * half;
            const int c = nt * 16 + l16;
            const float att = 1.0f / (1.0f + __expf(-acc[j]));   // sigmoid
            scale[(size_t)n * CC + g * CPART + c] = att * fw[n];
        }
    }
}

// -------------------------------------------------------------------------
// Pass 3: out = scale[n,c] * x. x is L2-resident from pass 1 (128MB < 192MB L2).
//  - x load:  nontemporal (last use; hit L2 but don't retain)
//  - out store: nontemporal (never re-read; don't let 128MB of output
//    write-allocate and evict x lines other blocks still need)
// One float4 per thread: b128 NT load + b128 NT store, fully coalesced.
// -------------------------------------------------------------------------
__global__ void apply_scale_kernel(const float* __restrict__ x,
                                   const float* __restrict__ scale,
                                   float* __restrict__ out) {
    const size_t i4 = (size_t)blockIdx.x * blockDim.x + threadIdx.x;
    const size_t nc = i4 >> 8;              // 256 float4 per (n,c) row
    const float  s  = scale[nc];

    const v4f* __restrict__ xin = (const v4f*)x;
    v4f*       __restrict__ o   = (v4f*)out;

    v4f v = __builtin_nontemporal_load(&xin[i4]);
    v *= s;
    __builtin_nontemporal_store(v, &o[i4]);
}

extern "C" void kernel_launch(void* const* d_in, const int* in_sizes, int n_in,
                              void* d_out, int out_size, void* d_ws, size_t ws_size,
                              hipStream_t stream) {
    const float* x  = (const float*)d_in[0];
    const float* W1 = (const float*)d_in[1];
    const float* W2 = (const float*)d_in[2];
    const float* M1 = (const float*)d_in[3];
    const float* M2 = (const float*)d_in[4];
    float* out = (float*)d_out;

    float* dct_freq = (float*)d_ws;             // 64*512 fp32 = 128 KB
    float* scale    = dct_freq + NN * CC;       // 64*512 fp32 = 128 KB

    // Pass 1: 64 n * 64 c-tiles = 4096 blocks, 8 waves each (one channel/wave)
    dct_pool_kernel<<<4096, 256, 0, stream>>>(x, dct_freq);

    // Pass 2: one block per group
    gating_kernel<<<NSPLIT, 256, 0, stream>>>(dct_freq, W1, W2, M1, M2, scale);

    // Pass 3: 64*512*1024/4 float4 = 8,388,608 -> 32768 blocks of 256
    apply_scale_kernel<<<32768, 256, 0, stream>>>(x, scale, out);
}